// Encoder_25864293057209
// MI455X (gfx1250) — compile-verified
//
#include <hip/hip_runtime.h>
#include <math.h>

typedef float v2f __attribute__((ext_vector_type(2)));
typedef float v8f __attribute__((ext_vector_type(8)));

#define N_NODES 50000
#define N_EDGES 800000
#define D 128
#define ACT_NONE 0
#define ACT_TANH 1
#define ACT_SILU 2

// ---------------------------------------------------------------------------
// In-degree (constant across layers: depends only on dst)
// ---------------------------------------------------------------------------
__global__ void degree_kernel(const int* __restrict__ dst, float* __restrict__ deg) {
    int e = blockIdx.x * blockDim.x + threadIdx.x;
    if (e < N_EDGES) atomicAdd(&deg[dst[e]], 1.0f);
}

// ---------------------------------------------------------------------------
// Scatter-add: agg[dst[e]] += h[src[e]]  (32 threads per edge, float4 each)
// ---------------------------------------------------------------------------
__global__ void scatter_kernel(const float* __restrict__ h, const int* __restrict__ src,
                               const int* __restrict__ dst, float* __restrict__ agg) {
    long tid = (long)blockIdx.x * blockDim.x + threadIdx.x;
    if (tid >= (long)N_EDGES * 32) return;
    int e  = (int)(tid >> 5);
    int c4 = (int)(tid & 31) << 2;
    int s = src[e], d = dst[e];
    float4 v = *(const float4*)(h + (long)s * D + c4);
    float* p = agg + (long)d * D + c4;
    atomicAdd(p + 0, v.x);
    atomicAdd(p + 1, v.y);
    atomicAdd(p + 2, v.z);
    atomicAdd(p + 3, v.w);
}

// ---------------------------------------------------------------------------
// Mean normalize: agg[n] /= max(deg[n], 1)
// ---------------------------------------------------------------------------
__global__ void normalize_kernel(float* __restrict__ agg, const float* __restrict__ deg) {
    long tid = (long)blockIdx.x * blockDim.x + threadIdx.x;
    if (tid >= (long)N_NODES * 32) return;
    int n  = (int)(tid >> 5);
    int c4 = (int)(tid & 31) << 2;
    float dv = 1.0f / fmaxf(deg[n], 1.0f);
    float4* p = (float4*)(agg + (long)n * D + c4);
    float4 v = *p;
    v.x *= dv; v.y *= dv; v.z *= dv; v.w *= dv;
    *p = v;
}

// ---------------------------------------------------------------------------
// Fused two-phase WMMA GEMM:
//   Out = act( A1 @ W1 + bias  [ + A2 @ W2 ] )
// Block: 256 threads = 8 wave32s. Each block -> 16 rows x 128 cols of Out.
// Each wave owns a 16x16 output tile via V_WMMA_F32_16X16X4_F32 (f32, K=4).
// W (128x128 = 64KB) + A tile (16x128 = 8KB) staged in LDS; the second phase
// barriers and reloads the SAME LDS so both products accumulate in the
// WMMA accumulator registers (no extra memory pass for the sum).
// ---------------------------------------------------------------------------
__launch_bounds__(256)
__global__ void sage_gemm_kernel(const float* __restrict__ A1, const float* __restrict__ W1,
                                 const float* __restrict__ bias,
                                 const float* __restrict__ A2, const float* __restrict__ W2,
                                 float* __restrict__ Out, int act) {
    __shared__ float sW[D * D];    // 64 KB
    __shared__ float sA[16 * D];   //  8 KB

    const int tid     = threadIdx.x;
    const int wave    = tid >> 5;
    const int lane    = tid & 31;
    const int rowBase = blockIdx.x * 16;
    const int colBase = wave * 16;
    const int lrow    = lane & 15;         // M (A) / N (B,C) index within tile
    const int hiHalf  = lane >> 4;         // lane-half selector
    const int koff    = hiHalf << 1;       // K offset within a K=4 step

    v8f c = {};                            // f32 16x16 accumulator (8 VGPRs)

    for (int phase = 0; phase < 2; ++phase) {
        const float* A = (phase == 0) ? A1 : A2;
        const float* W = (phase == 0) ? W1 : W2;
        if (A == nullptr) break;

        __syncthreads();   // protect LDS reuse between phases
        // cooperative load: W (4096 float4) and A tile (512 float4)
        for (int i = tid; i < (D * D) / 4; i += 256)
            ((float4*)sW)[i] = ((const float4*)W)[i];
        for (int i = tid; i < (16 * D) / 4; i += 256)
            ((float4*)sA)[i] = ((const float4*)(A + (long)rowBase * D))[i];
        __syncthreads();

        // K loop: 32 x V_WMMA_F32_16X16X4_F32
        // A 16x4: lane=M, VGPR v{0,1} = K {koff, koff+1}
        // B 4x16: lane=N, VGPR v{0,1} = K {koff, koff+1}
        #pragma unroll
        for (int k0 = 0; k0 < D; k0 += 4) {
            v2f a, b;
            a.x = sA[lrow * D + k0 + koff];
            a.y = sA[lrow * D + k0 + koff + 1];
            b.x = sW[(k0 + koff) * D + colBase + lrow];
            b.y = sW[(k0 + koff + 1) * D + colBase + lrow];
            c = __builtin_amdgcn_wmma_f32_16x16x4_f32(
                    /*neg_a=*/false, a, /*neg_b=*/false, b,
                    /*c_mod=*/(short)0, c, /*reuse_a=*/false, /*reuse_b=*/false);
        }
    }

    // Epilogue: C/D layout — VGPR r: lanes 0-15 -> M=r, lanes 16-31 -> M=r+8
    const int col = colBase + lrow;
    const float bv = bias ? bias[col] : 0.0f;
    #pragma unroll
    for (int r = 0; r < 8; ++r) {
        int grow = rowBase + r + (hiHalf << 3);
        float v = c[r] + bv;
        if (act == ACT_TANH)      v = tanhf(v);
        else if (act == ACT_SILU) v = v / (1.0f + __expf(-v));
        Out[(long)grow * D + col] = v;
    }
}

// ---------------------------------------------------------------------------
extern "C" void kernel_launch(void* const* d_in, const int* in_sizes, int n_in,
                              void* d_out, int out_size, void* d_ws, size_t ws_size,
                              hipStream_t stream) {
    const float* h       = (const float*)d_in[0];
    const int*   src     = (const int*)d_in[1];
    const int*   dst     = (const int*)d_in[2];
    const float* W_in1   = (const float*)d_in[3];
    const float* b_in1   = (const float*)d_in[4];
    const float* W_in2   = (const float*)d_in[5];
    const float* b_in2   = (const float*)d_in[6];
    const float* W_self  = (const float*)d_in[7];
    const float* b_self  = (const float*)d_in[8];
    const float* W_neigh = (const float*)d_in[9];
    float* out = (float*)d_out;

    const size_t ND = (size_t)N_NODES * D;
    float* buf0 = (float*)d_ws;    // ping
    float* buf1 = buf0 + ND;       // pong
    float* agg  = buf1 + ND;       // neighbor mean
    float* deg  = agg + ND;        // in-degree [N]

    // in-degree once (dst is the same for all layers)
    hipMemsetAsync(deg, 0, N_NODES * sizeof(float), stream);
    degree_kernel<<<(N_EDGES + 255) / 256, 256, 0, stream>>>(dst, deg);

    const int gemmGrid = N_NODES / 16;  // 3125, exact

    // fc_in: tanh(h@W1+b1) @ W2 + b2
    sage_gemm_kernel<<<gemmGrid, 256, 0, stream>>>(h,    W_in1, b_in1, nullptr, nullptr, buf0, ACT_TANH);
    sage_gemm_kernel<<<gemmGrid, 256, 0, stream>>>(buf0, W_in2, b_in2, nullptr, nullptr, buf1, ACT_NONE);

    float* hcur  = buf1;
    float* hnext = buf0;
    const long scatterThreads = (long)N_EDGES * 32;
    const long normThreads    = (long)N_NODES * 32;

    for (int i = 0; i < 3; ++i) {
        hipMemsetAsync(agg, 0, ND * sizeof(float), stream);
        scatter_kernel<<<(int)((scatterThreads + 255) / 256), 256, 0, stream>>>(hcur, src, dst, agg);
        normalize_kernel<<<(int)((normThreads + 255) / 256), 256, 0, stream>>>(agg, deg);

        float* o  = (i < 2) ? hnext : out;
        int   act = (i < 2) ? ACT_SILU : ACT_TANH;
        sage_gemm_kernel<<<gemmGrid, 256, 0, stream>>>(
            hcur, W_self + (size_t)i * D * D, b_self + (size_t)i * D,
            agg,  W_neigh + (size_t)i * D * D, o, act);

        float* t = hcur; hcur = o; hnext = t;
    }
}